// GraphAttentionNetwork_85572928406098
// MI455X (gfx1250) — compile-verified
//
#include <hip/hip_runtime.h>

// GAT (2-layer) for MI455X / gfx1250.
// N=100000 nodes, E=1600000 edges, NFEAT=256, NHID=64, NHEADS=4, NCLASS=40.
// GEMMs use V_WMMA_F32_16X16X4_F32 (exact fp32 matrix path, wave32).
// Edge softmax via order-preserving u32 keys + atomicMax, atomicAdd f32 scatter.

#define LRELU 0.2f

typedef __attribute__((ext_vector_type(2))) float v2f;
typedef __attribute__((ext_vector_type(8))) float v8f;

__device__ __forceinline__ float wred_sum(float v) {
#pragma unroll
    for (int o = 16; o > 0; o >>= 1) v += __shfl_xor(v, o, 32);
    return v;
}
__device__ __forceinline__ float wred_max(float v) {
#pragma unroll
    for (int o = 16; o > 0; o >>= 1) v = fmaxf(v, __shfl_xor(v, o, 32));
    return v;
}
// Order-preserving float -> u32 key (monotonic), for atomicMax-based segment max.
__device__ __forceinline__ unsigned fkey(float f) {
    unsigned u = __float_as_uint(f);
    return (u & 0x80000000u) ? ~u : (u | 0x80000000u);
}
__device__ __forceinline__ float fdec(unsigned k) {
    unsigned u = (k & 0x80000000u) ? (k & 0x7fffffffu) : ~k;
    return __uint_as_float(u);
}
__device__ __forceinline__ float lrelu(float x) { return x > 0.f ? x : LRELU * x; }

__global__ void fill_u32_kernel(unsigned* __restrict__ p, unsigned v, long long n) {
    long long i = (long long)blockIdx.x * blockDim.x + threadIdx.x;
    if (i < n) p[i] = v;
}

// ---------------------------------------------------------------------------
// WMMA f32 GEMM:  C[nrows x OUTW] = A[nrows x 256] * B
// MODE 0: B = W_heads (4,256,64) head-major; blockIdx.y = head strip of 64 cols,
//         C row stride 256, COLT=4.
// MODE 1: B = W_out (256,40) zero-padded to 48 cols; C row stride 40, COLT=3.
// Block = 256 threads (8 waves). Each block: 128 rows x (COLT*16) cols.
// Each wave: 16 rows, COLT accumulators (16x16 f32 tiles).
// B strip staged in LDS (<= 64KB), loaded coalesced once per block.
// ---------------------------------------------------------------------------
template <int COLT, int MODE>
__global__ __launch_bounds__(256) void gemm_wmma_f32(const float* __restrict__ A,
                                                     const float* __restrict__ B,
                                                     float* __restrict__ C, int nrows) {
    constexpr int CW = COLT * 16;
    __shared__ float Blds[256 * CW];
    const int tid = threadIdx.x;
    const int colBase = blockIdx.y * CW;

    for (int idx = tid; idx < 256 * CW; idx += 256) {
        float bv;
        if constexpr (MODE == 0) {
            // head = blockIdx.y; within-head (k,c) is exactly flat idx -> coalesced
            bv = B[(size_t)blockIdx.y * (256 * 64) + idx];
        } else {
            int k = idx / CW;
            int c = idx - k * CW;
            bv = (c < 40) ? B[k * 40 + c] : 0.0f;
        }
        Blds[idx] = bv;
    }
    __syncthreads();

    const int wid  = tid >> 5;
    const int lane = tid & 31;
    const int half = lane >> 4;   // 0: K 0..1, 1: K 2..3 (per 16x16x4 f32 layout)
    const int l16  = lane & 15;
    const int rowBase = blockIdx.x * 128 + wid * 16;

    int arow = rowBase + l16;
    if (arow >= nrows) arow = nrows - 1;   // clamp (stores are guarded)
    const float* aptr = A + (size_t)arow * 256;

    v8f acc[COLT] = {};

    for (int k0 = 0; k0 < 256; k0 += 4) {
        const int kb = k0 + half * 2;
        v2f a;
        a.x = aptr[kb];
        a.y = aptr[kb + 1];
#pragma unroll
        for (int t = 0; t < COLT; ++t) {
            v2f b;
            const int c = t * 16 + l16;
            b.x = Blds[kb * CW + c];
            b.y = Blds[(kb + 1) * CW + c];
            acc[t] = __builtin_amdgcn_wmma_f32_16x16x4_f32(false, a, false, b, (short)0,
                                                           acc[t], false, false);
        }
    }

    // Store. rowBase is wave-uniform: full tiles take a scalar-branch fast path
    // with no per-lane EXEC juggling.
    if (rowBase + 16 <= nrows) {
#pragma unroll
        for (int t = 0; t < COLT; ++t) {
            const int col = colBase + t * 16 + l16;
            if constexpr (MODE == 1) {
                if (col >= 40) continue;
            }
#pragma unroll
            for (int v = 0; v < 8; ++v) {
                const int row = rowBase + v + half * 8;
                if constexpr (MODE == 0) {
                    C[(size_t)row * 256 + col] = acc[t][v];
                } else {
                    C[(size_t)row * 40 + col] = acc[t][v];
                }
            }
        }
    } else {
#pragma unroll
        for (int t = 0; t < COLT; ++t) {
            const int col = colBase + t * 16 + l16;
#pragma unroll
            for (int v = 0; v < 8; ++v) {
                const int row = rowBase + v + half * 8;
                if constexpr (MODE == 0) {
                    if (row < nrows) C[(size_t)row * 256 + col] = acc[t][v];
                } else {
                    if (row < nrows && col < 40) C[(size_t)row * 40 + col] = acc[t][v];
                }
            }
        }
    }
}

// Per-node attention score dots, layer 1: one wave per node, 4 heads x 64 feats.
__global__ __launch_bounds__(256) void scores1_kernel(const float* __restrict__ Wh,
                                                      const float* __restrict__ a_heads,
                                                      float* __restrict__ ssrc,
                                                      float* __restrict__ sdst, int n) {
    long long gt = (long long)blockIdx.x * blockDim.x + threadIdx.x;
    int node = (int)(gt >> 5);
    int lane = threadIdx.x & 31;
    if (node >= n) return;
    const float* wh = Wh + (size_t)node * 256;
#pragma unroll
    for (int h = 0; h < 4; ++h) {
        float w1 = wh[h * 64 + lane];
        float w2 = wh[h * 64 + 32 + lane];
        const float* a = a_heads + h * 128;
        float ps = w1 * a[lane] + w2 * a[32 + lane];
        float pd = w1 * a[64 + lane] + w2 * a[96 + lane];
        ps = wred_sum(ps);
        pd = wred_sum(pd);
        if (lane == 0) {
            ssrc[node * 4 + h] = ps;
            sdst[node * 4 + h] = pd;
        }
    }
}

// Per-node attention score dots, layer 2: one wave per node, 40 feats.
__global__ __launch_bounds__(256) void scores2_kernel(const float* __restrict__ Wx,
                                                      const float* __restrict__ a_out,
                                                      float* __restrict__ ssrc,
                                                      float* __restrict__ sdst, int n) {
    long long gt = (long long)blockIdx.x * blockDim.x + threadIdx.x;
    int node = (int)(gt >> 5);
    int lane = threadIdx.x & 31;
    if (node >= n) return;
    const float* wx = Wx + (size_t)node * 40;
    float w1 = wx[lane];
    float w2 = (lane < 8) ? wx[32 + lane] : 0.f;
    float as2 = (lane < 8) ? a_out[32 + lane] : 0.f;
    float ad2 = (lane < 8) ? a_out[72 + lane] : 0.f;
    float ps = w1 * a_out[lane] + w2 * as2;
    float pd = w1 * a_out[40 + lane] + w2 * ad2;
    ps = wred_sum(ps);
    pd = wred_sum(pd);
    if (lane == 0) {
        ssrc[node] = ps;
        sdst[node] = pd;
    }
}

// ----- layer 1 edge passes: one thread per edge, 4 heads via b128 vector ops -----
__global__ __launch_bounds__(256) void edge_max1(const int* __restrict__ src,
                                                 const int* __restrict__ dst,
                                                 const float4* __restrict__ ss4,
                                                 const float4* __restrict__ sd4,
                                                 unsigned* __restrict__ mkey, int ne) {
    int e = blockIdx.x * blockDim.x + threadIdx.x;
    if (e >= ne) return;
    int s = src[e], d = dst[e];
    float4 vs = ss4[s];
    float4 vd = sd4[d];
    unsigned* mk = mkey + (size_t)d * 4;
    atomicMax(mk + 0, fkey(lrelu(vs.x + vd.x)));
    atomicMax(mk + 1, fkey(lrelu(vs.y + vd.y)));
    atomicMax(mk + 2, fkey(lrelu(vs.z + vd.z)));
    atomicMax(mk + 3, fkey(lrelu(vs.w + vd.w)));
}
__global__ __launch_bounds__(256) void edge_exp1(const int* __restrict__ src,
                                                 const int* __restrict__ dst,
                                                 const float4* __restrict__ ss4,
                                                 const float4* __restrict__ sd4,
                                                 const uint4* __restrict__ mkey4,
                                                 float* __restrict__ den,
                                                 float4* __restrict__ att4, int ne) {
    int e = blockIdx.x * blockDim.x + threadIdx.x;
    if (e >= ne) return;
    int s = src[e], d = dst[e];
    float4 vs = ss4[s];
    float4 vd = sd4[d];
    uint4 mk = mkey4[d];
    float4 ex;
    ex.x = expf(lrelu(vs.x + vd.x) - fdec(mk.x));
    ex.y = expf(lrelu(vs.y + vd.y) - fdec(mk.y));
    ex.z = expf(lrelu(vs.z + vd.z) - fdec(mk.z));
    ex.w = expf(lrelu(vs.w + vd.w) - fdec(mk.w));
    att4[e] = ex;
    float* dn = den + (size_t)d * 4;
    atomicAdd(dn + 0, ex.x);
    atomicAdd(dn + 1, ex.y);
    atomicAdd(dn + 2, ex.z);
    atomicAdd(dn + 3, ex.w);
}
__global__ __launch_bounds__(256) void edge_att1(const int* __restrict__ dst,
                                                 const float4* __restrict__ den4,
                                                 float4* __restrict__ att4, int ne) {
    int e = blockIdx.x * blockDim.x + threadIdx.x;
    if (e >= ne) return;
    float4 a = att4[e];
    float4 dn = den4[dst[e]];
    a.x /= dn.x;
    a.y /= dn.y;
    a.z /= dn.z;
    a.w /= dn.w;
    att4[e] = a;
}
// scatter: one thread per (edge, feature) ; feature 0..255, head = f>>6
__global__ __launch_bounds__(256) void scatter1(const int* __restrict__ src,
                                                const int* __restrict__ dst,
                                                const float* __restrict__ att,
                                                const float* __restrict__ Wh,
                                                float* __restrict__ hp, long long total) {
    long long i = (long long)blockIdx.x * blockDim.x + threadIdx.x;
    if (i >= total) return;
    int e = (int)(i >> 8);
    int f = (int)(i & 255);
    float a = att[(size_t)e * 4 + (f >> 6)];
    atomicAdd(&hp[(size_t)dst[e] * 256 + f], a * Wh[(size_t)src[e] * 256 + f]);
}
__global__ __launch_bounds__(256) void elu_kernel(float* __restrict__ p, long long n) {
    long long i = (long long)blockIdx.x * blockDim.x + threadIdx.x;
    if (i >= n) return;
    float v = p[i];
    p[i] = v > 0.f ? v : (expf(v) - 1.f);
}

// ----- layer 2 edge passes: one thread per edge -----
__global__ __launch_bounds__(256) void edge_max2(const int* __restrict__ src,
                                                 const int* __restrict__ dst,
                                                 const float* __restrict__ ss,
                                                 const float* __restrict__ sd,
                                                 unsigned* __restrict__ mkey, int ne) {
    int e = blockIdx.x * blockDim.x + threadIdx.x;
    if (e >= ne) return;
    float x = lrelu(ss[src[e]] + sd[dst[e]]);
    atomicMax(&mkey[dst[e]], fkey(x));
}
__global__ __launch_bounds__(256) void edge_exp2(const int* __restrict__ src,
                                                 const int* __restrict__ dst,
                                                 const float* __restrict__ ss,
                                                 const float* __restrict__ sd,
                                                 const unsigned* __restrict__ mkey,
                                                 float* __restrict__ den,
                                                 float* __restrict__ att, int ne) {
    int e = blockIdx.x * blockDim.x + threadIdx.x;
    if (e >= ne) return;
    int d = dst[e];
    float x = lrelu(ss[src[e]] + sd[d]);
    float ex = expf(x - fdec(mkey[d]));
    att[e] = ex;
    atomicAdd(&den[d], ex);
}
__global__ __launch_bounds__(256) void edge_att2(const int* __restrict__ dst,
                                                 const float* __restrict__ den,
                                                 float* __restrict__ att, int ne) {
    int e = blockIdx.x * blockDim.x + threadIdx.x;
    if (e >= ne) return;
    att[e] = att[e] / den[dst[e]];
}
// one thread per (edge, f<64), guard f<40
__global__ __launch_bounds__(256) void scatter2(const int* __restrict__ src,
                                                const int* __restrict__ dst,
                                                const float* __restrict__ att,
                                                const float* __restrict__ Wx,
                                                float* __restrict__ out, long long total) {
    long long i = (long long)blockIdx.x * blockDim.x + threadIdx.x;
    if (i >= total) return;
    int e = (int)(i >> 6);
    int f = (int)(i & 63);
    if (f >= 40) return;
    atomicAdd(&out[(size_t)dst[e] * 40 + f], att[e] * Wx[(size_t)src[e] * 40 + f]);
}

// log_softmax in place over 40 classes: one wave per node
__global__ __launch_bounds__(256) void logsoftmax_kernel(float* __restrict__ out, int n) {
    long long gt = (long long)blockIdx.x * blockDim.x + threadIdx.x;
    int node = (int)(gt >> 5);
    int lane = threadIdx.x & 31;
    if (node >= n) return;
    float* row = out + (size_t)node * 40;
    float v1 = row[lane];
    float v2 = (lane < 8) ? row[32 + lane] : -INFINITY;
    float mx = wred_max(fmaxf(v1, v2));
    float s = expf(v1 - mx) + ((lane < 8) ? expf(v2 - mx) : 0.f);
    s = wred_sum(s);
    float lg = mx + logf(s);
    row[lane] = v1 - lg;
    if (lane < 8) row[32 + lane] = v2 - lg;
}

extern "C" void kernel_launch(void* const* d_in, const int* in_sizes, int n_in, void* d_out,
                              int out_size, void* d_ws, size_t ws_size, hipStream_t stream) {
    (void)n_in; (void)out_size; (void)ws_size;
    const float* h      = (const float*)d_in[0];
    const int*   eidx   = (const int*)d_in[1];
    const float* Wheads = (const float*)d_in[2];
    const float* aheads = (const float*)d_in[3];
    const float* Wout   = (const float*)d_in[4];
    const float* aout   = (const float*)d_in[5];
    float* out = (float*)d_out;

    const int N = in_sizes[0] / 256;
    const int E = in_sizes[1] / 2;
    const int* src = eidx;
    const int* dst = eidx + E;

    float* ws = (float*)d_ws;
    // workspace layout (floats); all quad buffers stay 16B-aligned.
    size_t oWh   = 0;                          // N*256  layer-1 Wh (all heads)
    size_t oHP   = oWh  + (size_t)N * 256;     // N*256  h' -> x (elu in place)
    size_t oS1s  = oHP  + (size_t)N * 256;     // N*4
    size_t oS1d  = oS1s + (size_t)N * 4;       // N*4
    size_t oM1   = oS1d + (size_t)N * 4;       // N*4 (u32 keys)
    size_t oD1   = oM1  + (size_t)N * 4;       // N*4
    size_t oAtt1 = oD1  + (size_t)N * 4;       // E*4
    size_t oS2s  = oAtt1 + (size_t)E * 4;      // N
    size_t oS2d  = oS2s + (size_t)N;           // N
    size_t oM2   = oS2d + (size_t)N;           // N (u32 keys)
    size_t oD2   = oM2  + (size_t)N;           // N
    size_t oAtt2 = oD2  + (size_t)N;           // E
    size_t oWx   = oWh;                        // N*40, aliases Wh (dead by then)

    const unsigned NEG_INF_KEY = 0x007fffffu;  // fkey(-inf)

    // ---- init (every call: buffers are not re-poisoned between replays) ----
    hipMemsetAsync(ws + oHP, 0, (size_t)N * 256 * sizeof(float), stream);
    hipMemsetAsync(ws + oD1, 0, (size_t)N * 4 * sizeof(float), stream);
    hipMemsetAsync(ws + oD2, 0, (size_t)N * sizeof(float), stream);
    hipMemsetAsync(out, 0, (size_t)N * 40 * sizeof(float), stream);
    {
        long long n4 = (long long)N * 4;
        fill_u32_kernel<<<(int)((n4 + 255) / 256), 256, 0, stream>>>(
            (unsigned*)(ws + oM1), NEG_INF_KEY, n4);
        fill_u32_kernel<<<(N + 255) / 256, 256, 0, stream>>>(
            (unsigned*)(ws + oM2), NEG_INF_KEY, (long long)N);
    }

    // ---- layer 1 ----
    dim3 g1((N + 127) / 128, 4);
    gemm_wmma_f32<4, 0><<<g1, 256, 0, stream>>>(h, Wheads, ws + oWh, N);

    scores1_kernel<<<(N + 7) / 8, 256, 0, stream>>>(ws + oWh, aheads, ws + oS1s, ws + oS1d, N);

    int geE = (E + 255) / 256;
    edge_max1<<<geE, 256, 0, stream>>>(src, dst, (const float4*)(ws + oS1s),
                                       (const float4*)(ws + oS1d), (unsigned*)(ws + oM1), E);
    edge_exp1<<<geE, 256, 0, stream>>>(src, dst, (const float4*)(ws + oS1s),
                                       (const float4*)(ws + oS1d), (const uint4*)(ws + oM1),
                                       ws + oD1, (float4*)(ws + oAtt1), E);
    edge_att1<<<geE, 256, 0, stream>>>(dst, (const float4*)(ws + oD1), (float4*)(ws + oAtt1), E);

    long long tot1 = (long long)E * 256;
    scatter1<<<(int)((tot1 + 255) / 256), 256, 0, stream>>>(src, dst, ws + oAtt1, ws + oWh,
                                                            ws + oHP, tot1);
    long long nf = (long long)N * 256;
    elu_kernel<<<(int)((nf + 255) / 256), 256, 0, stream>>>(ws + oHP, nf);

    // ---- layer 2 ----
    dim3 g2((N + 127) / 128, 1);
    gemm_wmma_f32<3, 1><<<g2, 256, 0, stream>>>(ws + oHP, Wout, ws + oWx, N);

    scores2_kernel<<<(N + 7) / 8, 256, 0, stream>>>(ws + oWx, aout, ws + oS2s, ws + oS2d, N);

    edge_max2<<<geE, 256, 0, stream>>>(src, dst, ws + oS2s, ws + oS2d, (unsigned*)(ws + oM2), E);
    edge_exp2<<<geE, 256, 0, stream>>>(src, dst, ws + oS2s, ws + oS2d, (unsigned*)(ws + oM2),
                                       ws + oD2, ws + oAtt2, E);
    edge_att2<<<geE, 256, 0, stream>>>(dst, ws + oD2, ws + oAtt2, E);

    long long tot2 = (long long)E * 64;
    scatter2<<<(int)((tot2 + 255) / 256), 256, 0, stream>>>(src, dst, ws + oAtt2, ws + oWx, out,
                                                            tot2);

    logsoftmax_kernel<<<(N + 7) / 8, 256, 0, stream>>>(out, N);
}